// QuantumReservoir_40819369181744
// MI455X (gfx1250) — compile-verified
//
#include <hip/hip_runtime.h>
#include <math.h>

// ---------------------------------------------------------------------------
// QuantumReservoir on MI455X (gfx1250).
//
// Exact algebraic collapse: out[b, 3i+0] = 2(Cr[q]cos(x[b,q]) - Ci[q]sin(x[b,q]))
//                           out[b, 3i+1] = 2(Cr[q]sin(x[b,q]) + Ci[q]cos(x[b,q]))
//                           out[b, 3i+2] = Z[q],            q = 13 - i
// where Cr/Ci/Z are 42 batch-independent scalars derived from psi0.
//
// Kernel 1: one workgroup (1024 thr = 32 waves) simulates psi0 entirely in LDS
//   (128 KB interleaved float2 state; CDNA5's 320 KB/WGP makes this possible).
//   RX gates run on the matrix pipe: V_WMMA_F32_16X16X4_F32 with A = 4 stacked
//   copies of the real 4x4 RX matrix, B = 16 state pairs (one ds_load_b64 per
//   lane), full f32 precision.
// Kernel 2: 2048x14 readout (sincos + FMAs) -> 86016 output floats.
// ---------------------------------------------------------------------------

typedef __attribute__((ext_vector_type(2))) float v2f;
typedef __attribute__((ext_vector_type(8))) float v8f;

#define NQ 14
#define DIM 16384            // 2^14
#define PAIRS 8192           // DIM/2
#define NTROT 10
#define DT 0.1f              // EVO_TIME / N_TROTTER
#define NTHR 1024            // 32 waves
#define BATCH_NQ_THREADS 256

__global__ __launch_bounds__(NTHR)
void qr_base_state(const float* __restrict__ J, const float* __restrict__ g,
                   float* __restrict__ ws)
{
    extern __shared__ float smem[];
    float2* psi = (float2*)smem;          // [DIM] interleaved (re, im)
    float*  acc = smem + 2 * DIM;         // [48] (42 used)

    const int tid  = threadIdx.x;
    const int lane = tid & 31;
    const int wave = tid >> 5;

    // |psi0> after 14 Hadamards on |0...0>: uniform +1/128.
    for (int k = tid; k < DIM; k += NTHR) psi[k] = make_float2(0.0078125f, 0.0f);
    if (tid < 48) acc[tid] = 0.0f;

    // Fused ZZ diagonal for one Trotter step: angle(k) = sum_i (+-) J[i]*dt/2,
    // + if bits i,i+1 differ, - if equal. Identical every step -> cache
    // cos/sin per thread-owned element (16 elements/thread) in registers.
    float Jl[NQ - 1];
#pragma unroll
    for (int i = 0; i < NQ - 1; ++i) Jl[i] = 0.5f * DT * J[i];

    float zc[16], zs[16];
#pragma unroll
    for (int j = 0; j < 16; ++j) {
        int k = tid + j * NTHR;
        float ang = 0.0f;
#pragma unroll
        for (int i = 0; i < NQ - 1; ++i) {
            int d = ((k >> i) ^ (k >> (i + 1))) & 1;   // bits differ?
            ang += d ? Jl[i] : -Jl[i];
        }
        sincosf(ang, &zs[j], &zc[j]);
    }

    // Per-layer WMMA A-operand values, hoisted out of the Trotter loop.
    // M4 rows (components [p0re p0im p1re p1im]):
    //  r0: [ c  0  0  s]   r1: [ 0  c -s  0]
    //  r2: [ 0  s  c  0]   r3: [-s  0  0  c]
    // A(16x4) = 4 stacked copies of M4. Lane L<16 supplies A[L][0],A[L][1];
    // lane L>=16 supplies A[L-16][2],A[L-16][3].
    const bool hi = (lane >= 16);
    const int  m  = lane & 3;
    float aX[NQ], aY[NQ];
#pragma unroll
    for (int q = 0; q < NQ; ++q) {
        float c, s;
        sincosf(0.5f * g[q] * DT, &s, &c);
        aX[q] = hi ? ((m == 1) ? -s : ((m == 2) ? c : 0.0f))
                   : ((m == 0) ?  c : ((m == 3) ? -s : 0.0f));
        aY[q] = hi ? ((m == 0) ?  s : ((m == 3) ? c : 0.0f))
                   : ((m == 1) ?  c : ((m == 2) ? s : 0.0f));
    }
    __syncthreads();

    for (int t = 0; t < NTROT; ++t) {
        // ---- all 13 ZZ rotations fused: one diagonal complex multiply ----
#pragma unroll
        for (int j = 0; j < 16; ++j) {
            int k = tid + j * NTHR;
            float2 p = psi[k];
            psi[k] = make_float2(p.x * zc[j] - p.y * zs[j],
                                 p.x * zs[j] + p.y * zc[j]);
        }
        __syncthreads();

        // ---- RX(q) layers via V_WMMA_F32_16X16X4_F32 ----
        for (int q = 0; q < NQ; ++q) {
            v2f a; a.x = aX[q]; a.y = aY[q];
            const int lo = 1 << q, lm = lo - 1;
            // Each wave owns 256 pairs -> 16 WMMAs of 16 pairs each.
            for (int chunk = 0; chunk < 16; ++chunk) {
                int n  = wave * 256 + chunk * 16 + (lane & 15);
                int k0 = ((n >> q) << (q + 1)) | (n & lm);
                int k  = hi ? (k0 | lo) : k0;
                // B(4x16): lanes 0-15 hold (p0re,p0im), lanes 16-31 (p1re,p1im):
                // one ds_load_b64 per lane.
                float2 p = psi[k];
                v2f b; b.x = p.x; b.y = p.y;
                v8f cz = {};
                v8f d = __builtin_amdgcn_wmma_f32_16x16x4_f32(
                            false, a, false, b, (short)0, cz, false, false);
                // D rows 0..3 (lanes<16) / 8..11 (lanes>=16) hold this pair's
                // updated components; one ds_store_b64 per lane.
                float nr = hi ? d[2] : d[0];
                float ni = hi ? d[3] : d[1];
                psi[k] = make_float2(nr, ni);
            }
            __syncthreads();
        }
    }

    // ---- reductions: C(q) = sum conj(p0)*p1, Z(q) = sum |p0|^2-|p1|^2 ----
    for (int q = 0; q < NQ; ++q) {
        const int lo = 1 << q, lm = lo - 1;
        float cr = 0.0f, ci = 0.0f, zz = 0.0f;
        for (int p = tid; p < PAIRS; p += NTHR) {
            int k0 = ((p >> q) << (q + 1)) | (p & lm);
            float2 p0 = psi[k0];
            float2 p1 = psi[k0 | lo];
            cr += p0.x * p1.x + p0.y * p1.y;                     // Re(conj(p0)*p1)
            ci += p0.x * p1.y - p0.y * p1.x;                     // Im(conj(p0)*p1)
            zz += p0.x * p0.x + p0.y * p0.y - p1.x * p1.x - p1.y * p1.y;
        }
#pragma unroll
        for (int off = 16; off > 0; off >>= 1) {
            cr += __shfl_xor(cr, off, 32);
            ci += __shfl_xor(ci, off, 32);
            zz += __shfl_xor(zz, off, 32);
        }
        if (lane == 0) {
            atomicAdd(&acc[q],          cr);
            atomicAdd(&acc[NQ + q],     ci);
            atomicAdd(&acc[2 * NQ + q], zz);
        }
    }
    __syncthreads();
    if (tid < 3 * NQ) ws[tid] = acc[tid];
}

__global__ __launch_bounds__(BATCH_NQ_THREADS)
void qr_readout(const float* __restrict__ x, const float* __restrict__ ws,
                float* __restrict__ out, int batch)
{
    int gid = blockIdx.x * blockDim.x + threadIdx.x;
    if (gid >= batch * NQ) return;
    int b = gid / NQ;
    int i = gid - b * NQ;
    int q = NQ - 1 - i;
    float xq = x[b * NQ + q];
    float cr = ws[q], ci = ws[NQ + q], zz = ws[2 * NQ + q];
    float sn, cs;
    sincosf(xq, &sn, &cs);
    // 3 consecutive floats -> backend can merge into one global_store_b96.
    float* o = out + b * (3 * NQ) + 3 * i;
    o[0] = 2.0f * (cr * cs - ci * sn);   // 2 Re(e^{ix} C)
    o[1] = 2.0f * (cr * sn + ci * cs);   // 2 Im(e^{ix} C)
    o[2] = zz;
}

extern "C" void kernel_launch(void* const* d_in, const int* in_sizes, int n_in,
                              void* d_out, int out_size, void* d_ws, size_t ws_size,
                              hipStream_t stream)
{
    const float* x = (const float*)d_in[0];   // (BATCH, 14)
    const float* J = (const float*)d_in[1];   // (13,)
    const float* g = (const float*)d_in[2];   // (14,)
    float* out = (float*)d_out;               // (BATCH, 42)
    float* ws  = (float*)d_ws;                // 42 floats of scratch

    const int batch = in_sizes[0] / NQ;

    // 2*DIM state floats + 48 accumulator floats = ~128.2 KB LDS (320 KB/WGP).
    size_t smem = (size_t)(2 * DIM + 48) * sizeof(float);
    qr_base_state<<<1, NTHR, smem, stream>>>(J, g, ws);

    int total  = batch * NQ;
    int blocks = (total + BATCH_NQ_THREADS - 1) / BATCH_NQ_THREADS;
    qr_readout<<<blocks, BATCH_NQ_THREADS, 0, stream>>>(x, ws, out, batch);
}